// NormalizedGCNConv_4827543240746
// MI455X (gfx1250) — compile-verified
//
#include <hip/hip_runtime.h>

#define IN_DIM   256
#define OUT_DIM  128
#define SCALE_F  1.8f
#define ALPHA_F  0.15f
#define EPS_F    1e-12f

typedef float v2f __attribute__((ext_vector_type(2)));
typedef float v8f __attribute__((ext_vector_type(8)));

// ---------------------------------------------------------------------------
// Kernel 1: h = L2normalize_row(x @ W^T + b) * SCALE
// One wave per 16-row M-tile; wave computes all 128 output columns (8 N-tiles)
// with V_WMMA_F32_16X16X4_F32, then fuses bias + row-norm + scale in-register.
// Requires N % 16 == 0 (true: 50000 = 3125 * 16).
// ---------------------------------------------------------------------------
__global__ __launch_bounds__(128) void gemm_norm_kernel(
    const float* __restrict__ x, const float* __restrict__ W,
    const float* __restrict__ bias, float* __restrict__ h, int nTilesM)
{
    const int lane = threadIdx.x & 31;
    const int wave = threadIdx.x >> 5;
    const int tileM = blockIdx.x * 4 + wave;
    if (tileM >= nTilesM) return;                 // wave-uniform exit

    const int m  = lane & 15;                     // A: row within tile / B: col
    const int hi = lane >> 4;                     // half-wave selects K pair

    const float* xr = x + (size_t)(tileM * 16 + m) * IN_DIM + 2 * hi;

    v8f acc[8];
    #pragma unroll
    for (int t = 0; t < 8; ++t) acc[t] = (v8f){};

    // K loop: 64 chunks of K=4
    for (int k = 0; k < IN_DIM; k += 4) {
        // A lane layout (32-bit A 16x4): lane holds A[m][k + 2*hi + {0,1}]
        v2f a = *(const v2f*)(xr + k);
        #pragma unroll
        for (int t = 0; t < 8; ++t) {
            // B lane layout mirrors A: lane holds B[k + 2*hi + {0,1}][n],
            // B[kk][n] = W[n][kk]  (n = t*16 + m)
            v2f bb = *(const v2f*)(W + (size_t)(t * 16 + m) * IN_DIM + k + 2 * hi);
            acc[t] = __builtin_amdgcn_wmma_f32_16x16x4_f32(
                false, a, false, bb, (short)0, acc[t], false, false);
        }
    }

    // Bias: C/D element (t, i) sits at row M = i + 8*hi, col = t*16 + m
    #pragma unroll
    for (int t = 0; t < 8; ++t) {
        float bv = bias[t * 16 + m];
        #pragma unroll
        for (int i = 0; i < 8; ++i) acc[t][i] += bv;
    }

    // Row L2 norm: per VGPR index i, reduce sum(h^2) across 16 lanes of each
    // half-wave (xor masks 1,2,4,8 never cross the 16-lane boundary).
    const size_t rowBase = (size_t)(tileM * 16 + 8 * hi) * OUT_DIM;
    #pragma unroll
    for (int i = 0; i < 8; ++i) {
        float s = 0.f;
        #pragma unroll
        for (int t = 0; t < 8; ++t) s += acc[t][i] * acc[t][i];
        s += __shfl_xor(s, 1);
        s += __shfl_xor(s, 2);
        s += __shfl_xor(s, 4);
        s += __shfl_xor(s, 8);
        float sc = SCALE_F / fmaxf(sqrtf(s), EPS_F);
        #pragma unroll
        for (int t = 0; t < 8; ++t) acc[t][i] *= sc;
        // store row (M = i + 8*hi): consecutive lanes -> consecutive cols
        #pragma unroll
        for (int t = 0; t < 8; ++t)
            h[rowBase + (size_t)i * OUT_DIM + t * 16 + m] = acc[t][i];
    }
}

// ---------------------------------------------------------------------------
// Degree / dinv kernels (deg buffer doubles as dinv after rsqrt)
// ---------------------------------------------------------------------------
__global__ void deg_init_kernel(float* __restrict__ deg, int n) {
    int i = blockIdx.x * blockDim.x + threadIdx.x;
    if (i < n) deg[i] = 1.0f;                      // self loop
}

__global__ void deg_acc_kernel(const int* __restrict__ col,
                               float* __restrict__ deg, int e) {
    int i = blockIdx.x * blockDim.x + threadIdx.x;
    if (i < e) atomicAdd(&deg[col[i]], 1.0f);
}

__global__ void dinv_kernel(float* __restrict__ deg, int n) {
    int i = blockIdx.x * blockDim.x + threadIdx.x;
    if (i < n) deg[i] = __frsqrt_rn(deg[i]);       // deg >= 1 always
}

// ---------------------------------------------------------------------------
// agg[i][:] = dinv[i]^2 * z[i][:]   (self-loop term, atomics-free init)
// Indexed by float4 quads: nquads = N * 32
// ---------------------------------------------------------------------------
__global__ void agg_init_kernel(const float* __restrict__ z,
                                const float* __restrict__ dinv,
                                float* __restrict__ A, int nquads) {
    int i = blockIdx.x * blockDim.x + threadIdx.x;
    if (i >= nquads) return;
    int r = i >> 5;                                // 32 quads per row
    float d = dinv[r];
    float w = d * d;
    float4 v = ((const float4*)z)[i];
    float4 o;
    o.x = w * v.x; o.y = w * v.y; o.z = w * v.z; o.w = w * v.w;
    ((float4*)A)[i] = o;
}

// ---------------------------------------------------------------------------
// Edge scatter: one wave per edge; each lane handles 4 features.
// agg[col][:] += dinv[row]*dinv[col] * z[row][:]
// ---------------------------------------------------------------------------
__global__ __launch_bounds__(256) void scatter_kernel(
    const int* __restrict__ rowIdx, const int* __restrict__ colIdx,
    const float* __restrict__ dinv, const float* __restrict__ z,
    float* __restrict__ A, int E)
{
    int wave = blockIdx.x * (blockDim.x >> 5) + (threadIdx.x >> 5);
    int lane = threadIdx.x & 31;
    if (wave >= E) return;
    int r = rowIdx[wave];
    int c = colIdx[wave];
    float w = dinv[r] * dinv[c];
    float4 v = ((const float4*)(z + (size_t)r * OUT_DIM))[lane];
    float* dst = A + (size_t)c * OUT_DIM + lane * 4;
    atomicAdd(dst + 0, w * v.x);                   // non-returning f32 atomics
    atomicAdd(dst + 1, w * v.y);
    atomicAdd(dst + 2, w * v.z);
    atomicAdd(dst + 3, w * v.w);
}

// ---------------------------------------------------------------------------
// z' = (1-ALPHA)*agg + ALPHA*h
// ---------------------------------------------------------------------------
__global__ void combine_kernel(const float* __restrict__ A,
                               const float* __restrict__ h,
                               float* __restrict__ z, int nquads) {
    int i = blockIdx.x * blockDim.x + threadIdx.x;
    if (i >= nquads) return;
    float4 a = ((const float4*)A)[i];
    float4 b = ((const float4*)h)[i];
    float4 o;
    o.x = (1.0f - ALPHA_F) * a.x + ALPHA_F * b.x;
    o.y = (1.0f - ALPHA_F) * a.y + ALPHA_F * b.y;
    o.z = (1.0f - ALPHA_F) * a.z + ALPHA_F * b.z;
    o.w = (1.0f - ALPHA_F) * a.w + ALPHA_F * b.w;
    ((float4*)z)[i] = o;
}

// ---------------------------------------------------------------------------
extern "C" void kernel_launch(void* const* d_in, const int* in_sizes, int n_in,
                              void* d_out, int out_size, void* d_ws, size_t ws_size,
                              hipStream_t stream) {
    const float* x    = (const float*)d_in[0];
    const int*   eidx = (const int*)d_in[1];     // [2, E] flat: row then col
    const float* W    = (const float*)d_in[2];   // [128, 256]
    const float* bias = (const float*)d_in[3];   // [128]
    float* out = (float*)d_out;                  // [N, 128]

    const int N = in_sizes[0] / IN_DIM;
    const int E = in_sizes[1] / 2;
    const int* erow = eidx;
    const int* ecol = eidx + E;

    // workspace: h [N*128] | agg [N*128] | dinv [N]   (~51.4 MB)
    float* h    = (float*)d_ws;
    float* A    = h + (size_t)N * OUT_DIM;
    float* dinv = A + (size_t)N * OUT_DIM;

    const int tb = 256;
    const int nTilesM = (N + 15) / 16;
    const int nquads = N * (OUT_DIM / 4);

    // 1) h = normalize(x@W^T + b) * SCALE   (WMMA f32)
    gemm_norm_kernel<<<(nTilesM + 3) / 4, 128, 0, stream>>>(x, W, bias, h, nTilesM);

    // 2) symmetric GCN normalization weights
    deg_init_kernel<<<(N + tb - 1) / tb, tb, 0, stream>>>(dinv, N);
    deg_acc_kernel <<<(E + tb - 1) / tb, tb, 0, stream>>>(ecol, dinv, E);
    dinv_kernel    <<<(N + tb - 1) / tb, tb, 0, stream>>>(dinv, N);

    // 3) APPNP iteration 1 (z = h), z1 -> d_out
    agg_init_kernel<<<(nquads + tb - 1) / tb, tb, 0, stream>>>(h, dinv, A, nquads);
    scatter_kernel <<<(E + 7) / 8, 256, 0, stream>>>(erow, ecol, dinv, h, A, E);
    combine_kernel <<<(nquads + tb - 1) / tb, tb, 0, stream>>>(A, h, out, nquads);

    // 4) APPNP iteration 2 (z = d_out), result -> d_out
    agg_init_kernel<<<(nquads + tb - 1) / tb, tb, 0, stream>>>(out, dinv, A, nquads);
    scatter_kernel <<<(E + 7) / 8, 256, 0, stream>>>(erow, ecol, dinv, out, A, E);
    combine_kernel <<<(nquads + tb - 1) / tb, tb, 0, stream>>>(A, h, out, nquads);
}